// QuantumLayer_59906203844853
// MI455X (gfx1250) — compile-verified
//
#include <hip/hip_runtime.h>
#include <hip/hip_bf16.h>

typedef __attribute__((ext_vector_type(2))) float v2f;
typedef __attribute__((ext_vector_type(8))) float v8f;

#define NQ      12
#define DIM     4096
#define LAYERS  4
#define TPB     256
#define FEATS_PER 36

// State stored interleaved {re,im} (v2f). Padding: +1 slot per 16 -> stride-16 group-2
// accesses become stride-17 (34 floats): n*34 mod 64 banks is injective over 16 lanes.
__device__ __forceinline__ int sp(int i) { return i + (i >> 4); }

// (row, col) -> raw state index for tile t of group g.
// g0: top 4 bits (wires 0-3); g1: middle 4 bits (wires 4-7); g2: low 4 bits (wires 8-11,
// gate applied on the right via transpose trick: C'(i,c) = sum_j U[i][j] * S[(16t+c)*16 + j]).
__device__ __forceinline__ int goff(int g, int t, int row, int col) {
  if (g == 0) return row * 256 + t * 16 + col;
  if (g == 1) return t * 256 + row * 16 + col;
  return (t * 16 + col) * 16 + row;
}

__device__ __forceinline__ v8f wmma_f32(v2f a, v2f b, v8f c) {
  // (neg_a, A, neg_b, B, c_mod, C, reuse_a, reuse_b); f32 A/B neg must be 0 per ISA -> negate in regs.
  return __builtin_amdgcn_wmma_f32_16x16x4_f32(false, a, false, b, (short)0, c, false, false);
}

// Single-wave complex 16x16x16 matmul on planar LDS matrices: D = A x B.
// A(m,k)=a_[m*16+k], B(k,n)=b_[k*16+n], D(m,n)=d_[m*16+n]. Wave-uniform call, EXEC all ones.
__device__ __forceinline__ void mm16_wave(float* dRe, float* dIm,
                                          const float* aRe_, const float* aIm_,
                                          const float* bRe_, const float* bIm_,
                                          int lane) {
  const int n = lane & 15, hi = lane >> 4;
  v2f aRe[4], aIm[4], aImN[4], bRe[4], bIm[4];
#pragma unroll
  for (int kk = 0; kk < 4; ++kk) {
    int k0 = kk * 4 + 2 * hi;
    v2f r, q, qn, br, bi;
    r[0] = aRe_[n * 16 + k0];      r[1] = aRe_[n * 16 + k0 + 1];
    q[0] = aIm_[n * 16 + k0];      q[1] = aIm_[n * 16 + k0 + 1];
    qn[0] = -q[0];                 qn[1] = -q[1];
    br[0] = bRe_[k0 * 16 + n];     br[1] = bRe_[(k0 + 1) * 16 + n];
    bi[0] = bIm_[k0 * 16 + n];     bi[1] = bIm_[(k0 + 1) * 16 + n];
    aRe[kk] = r; aIm[kk] = q; aImN[kk] = qn; bRe[kk] = br; bIm[kk] = bi;
  }
  v8f cRe = {}, cIm = {};
#pragma unroll
  for (int kk = 0; kk < 4; ++kk) cRe = wmma_f32(aRe[kk],  bRe[kk], cRe);
#pragma unroll
  for (int kk = 0; kk < 4; ++kk) cRe = wmma_f32(aImN[kk], bIm[kk], cRe);
#pragma unroll
  for (int kk = 0; kk < 4; ++kk) cIm = wmma_f32(aRe[kk],  bIm[kk], cIm);
#pragma unroll
  for (int kk = 0; kk < 4; ++kk) cIm = wmma_f32(aIm[kk],  bRe[kk], cIm);
#pragma unroll
  for (int v = 0; v < 8; ++v) {
    dRe[(v + 8 * hi) * 16 + n] = cRe[v];
    dIm[(v + 8 * hi) * 16 + n] = cIm[v];
  }
}

// Full-state pass: apply 16x16 complex unitary (gr,gi planar; may be wave-selected) along
// group g's index. Tiles t = wave + 8*tt, so both tiles of a wave share parity (wave&1).
// fold: pre-apply diagonal ZZ(7,8) phase to inputs; sign from (col&1)^(k>>3).
__device__ __forceinline__ void apply_group(v2f* sAmp, const float* gr, const float* gi,
                                            int g, int tid, bool fold, float zc, float zs) {
  const int wave = tid >> 5;
  const int lane = tid & 31;
  const int n    = lane & 15;   // A: M index; B/D: column index
  const int hi   = lane >> 4;   // lane half selects K pair / D row half

  // A (gate): 16x4 f32 layout. Lane m = lane&15; VGPR0 K = 4*kk+2*hi, VGPR1 K = 4*kk+2*hi+1.
  v2f aRe[4], aIm[4], aImN[4];
#pragma unroll
  for (int kk = 0; kk < 4; ++kk) {
    int k0 = kk * 4 + 2 * hi;
    v2f r, q, qn;
    r[0] = gr[n * 16 + k0];   r[1] = gr[n * 16 + k0 + 1];
    q[0] = gi[n * 16 + k0];   q[1] = gi[n * 16 + k0 + 1];
    qn[0] = -q[0];            qn[1] = -q[1];
    aRe[kk] = r; aIm[kk] = q; aImN[kk] = qn;
  }

#pragma unroll
  for (int tt = 0; tt < 2; ++tt) {
    int t = wave + 8 * tt;
    v2f bRe[4], bIm[4];
#pragma unroll
    for (int kk = 0; kk < 4; ++kk) {
      int k0 = kk * 4 + 2 * hi;
      v2f e0 = sAmp[sp(goff(g, t, k0,     n))];  // one ds_load_b64
      v2f e1 = sAmp[sp(goff(g, t, k0 + 1, n))];
      if (fold) {
        float s0 = ((n ^ (k0 >> 3)) & 1) ? -zs : zs;          // equal bits -> e^{-it/2}
        float s1 = ((n ^ ((k0 + 1) >> 3)) & 1) ? -zs : zs;
        v2f f0, f1;
        f0[0] = zc * e0[0] + s0 * e0[1];  f0[1] = zc * e0[1] - s0 * e0[0];
        f1[0] = zc * e1[0] + s1 * e1[1];  f1[1] = zc * e1[1] - s1 * e1[0];
        e0 = f0; e1 = f1;
      }
      v2f br, bi;
      br[0] = e0[0]; br[1] = e1[0];
      bi[0] = e0[1]; bi[1] = e1[1];
      bRe[kk] = br; bIm[kk] = bi;
    }
    v8f cRe = {}; v8f cIm = {};
#pragma unroll
    for (int kk = 0; kk < 4; ++kk) cRe = wmma_f32(aRe[kk],  bRe[kk], cRe);
#pragma unroll
    for (int kk = 0; kk < 4; ++kk) cRe = wmma_f32(aImN[kk], bIm[kk], cRe);
#pragma unroll
    for (int kk = 0; kk < 4; ++kk) cIm = wmma_f32(aRe[kk],  bIm[kk], cIm);
#pragma unroll
    for (int kk = 0; kk < 4; ++kk) cIm = wmma_f32(aIm[kk],  bRe[kk], cIm);
#pragma unroll
    for (int v = 0; v < 8; ++v) {
      v2f o; o[0] = cRe[v]; o[1] = cIm[v];
      sAmp[sp(goff(g, t, v + 8 * hi, n))] = o;   // one ds_store_b64
    }
  }
}

__global__ __launch_bounds__(TPB) void qsim_kernel(const float* __restrict__ feats,
                                                   const float* __restrict__ wrot,
                                                   const float* __restrict__ wenta,
                                                   float* __restrict__ out) {
  __shared__ v2f   sAmp[DIM + DIM / 16];          // interleaved state, padded
  __shared__ float KRe[3 * 256], KIm[3 * 256];    // per-sample kron unitaries (layer-invariant)
  __shared__ float MRe[3 * 256], MIm[3 * 256];    // micro outputs: M0=G0c*K0, M1=G1c, M2=G2c
  __shared__ float TRe[2 * 256], TIm[2 * 256];    // T_par = D_par(ZZ34) * K1
  __shared__ float P1Re[2 * 256], P1Im[2 * 256];  // M1_par = G1c * T_par
  __shared__ float P2Re[256],     P2Im[256];      // K2 * G2c (last layer: G2c)
  __shared__ float uRe[NQ * 4],  uIm[NQ * 4];
  __shared__ float rotC[NQ], rotS[NQ], entC[NQ], entS[NQ];
  __shared__ float red[8 * NQ];

  const int tid  = threadIdx.x;
  const int blk  = blockIdx.x;
  const int wave = tid >> 5;
  const int lane = tid & 31;
  const int mr   = tid >> 4;   // matrix row for 16x16 work
  const int mc   = tid & 15;   // matrix col

  __builtin_prefetch(feats + blk * FEATS_PER, 0, 0);   // global_prefetch_b8

  // Rot(f0,f1,f2) = RZ(f2) RY(f1) RZ(f0):
  //   u00 = c e^{-ia}, u01 = -s e^{ib}, u10 = s e^{-ib}, u11 = c e^{ia}; a=(f0+f2)/2, b=(f0-f2)/2
  if (tid < NQ) {
    float f0 = feats[blk * FEATS_PER + tid * 3 + 0];
    float f1 = feats[blk * FEATS_PER + tid * 3 + 1];
    float f2 = feats[blk * FEATS_PER + tid * 3 + 2];
    float c = cosf(0.5f * f1), s = sinf(0.5f * f1);
    float a = 0.5f * (f0 + f2), b = 0.5f * (f0 - f2);
    float ca = cosf(a), sa = sinf(a), cb = cosf(b), sb = sinf(b);
    uRe[tid * 4 + 0] =  c * ca;  uIm[tid * 4 + 0] = -c * sa;
    uRe[tid * 4 + 1] = -s * cb;  uIm[tid * 4 + 1] = -s * sb;
    uRe[tid * 4 + 2] =  s * cb;  uIm[tid * 4 + 2] = -s * sb;
    uRe[tid * 4 + 3] =  c * ca;  uIm[tid * 4 + 3] =  c * sa;
  }
  __syncthreads();

  // K_g = u_{4g} (x) u_{4g+1} (x) u_{4g+2} (x) u_{4g+3}   (wire 4g = group MSB)
#pragma unroll
  for (int g = 0; g < 3; ++g) {
    float pr = 1.0f, pi = 0.0f;
#pragma unroll
    for (int b = 0; b < 4; ++b) {
      int q  = 4 * g + b;
      int ib = (mr >> (3 - b)) & 1;
      int jb = (mc >> (3 - b)) & 1;
      float ur = uRe[q * 4 + ib * 2 + jb];
      float ui = uIm[q * 4 + ib * 2 + jb];
      float nr = pr * ur - pi * ui;
      float ni = pr * ui + pi * ur;
      pr = nr; pi = ni;
    }
    KRe[g * 256 + tid] = pr;
    KIm[g * 256 + tid] = pi;
  }
  __syncthreads();

  // state = U2 |0...0>  ==  column 0 of K2 in the low-16 slots (U0 lives in M0, U1 in M1_par).
#pragma unroll
  for (int k = 0; k < DIM / TPB; ++k) {
    int idx = tid + TPB * k;
    v2f v; v[0] = 0.0f; v[1] = 0.0f;
    if (idx < 16) { v[0] = KRe[512 + idx * 16]; v[1] = KIm[512 + idx * 16]; }
    sAmp[sp(idx)] = v;
  }

  for (int l = 0; l < LAYERS; ++l) {
    // Per-layer shared-gate scalars
    if (tid < NQ) {
      float w = wrot[l * NQ + tid];
      rotC[tid] = cosf(0.5f * w);
      rotS[tid] = sinf(0.5f * w);
    } else if (tid >= 32 && tid < 32 + (NQ - 1)) {
      int j = tid - 32;
      float w = wenta[l * (NQ - 1) + j];
      entC[j] = cosf(0.5f * w);
      entS[j] = sinf(0.5f * w);
    }
    __syncthreads();

    // Init micro matrices (M0 = K0, M1 = M2 = I) and T_par = D_par(ZZ34) * K1 (row phases).
    {
      float idv = (mr == mc) ? 1.0f : 0.0f;
      MRe[tid]       = KRe[tid];  MIm[tid]       = KIm[tid];
      MRe[256 + tid] = idv;       MIm[256 + tid] = 0.0f;
      MRe[512 + tid] = idv;       MIm[512 + tid] = 0.0f;
      float zc = entC[3], zs = entS[3];
      float k1r = KRe[256 + tid], k1i = KIm[256 + tid];
      int rb = mr >> 3;
      float sg0 = (rb == 0) ? zs : -zs;
      float sg1 = (rb == 1) ? zs : -zs;
      TRe[tid]       = zc * k1r + sg0 * k1i;  TIm[tid]       = zc * k1i - sg0 * k1r;
      TRe[256 + tid] = zc * k1r + sg1 * k1i;  TIm[256 + tid] = zc * k1i - sg1 * k1r;
    }
    __syncthreads();

    // 7 micro-gates (identical structure per group): left-multiply the three 16x16 matrices.
    // Sequence: RX(b0), XX(b0,b1), RY(b1), ZZ(b1,b2), RX(b2), XX(b2,b3), RY(b3); bitpos p = 3-b.
    auto rot_step = [&](int b, bool isRX) {
      int p  = 3 - b;
      int rp = mr ^ (1 << p);
      int bit = (mr >> p) & 1;
      float orv[3], oiv[3], prv[3], piv[3];
#pragma unroll
      for (int g = 0; g < 3; ++g) {
        orv[g] = MRe[g * 256 + mr * 16 + mc];  oiv[g] = MIm[g * 256 + mr * 16 + mc];
        prv[g] = MRe[g * 256 + rp * 16 + mc];  piv[g] = MIm[g * 256 + rp * 16 + mc];
      }
      __syncthreads();
#pragma unroll
      for (int g = 0; g < 3; ++g) {
        float cc = rotC[4 * g + b], ss = rotS[4 * g + b];
        float nr, ni;
        if (isRX) {                      // [[c,-is],[-is,c]] : symmetric in bit
          nr = cc * orv[g] + ss * piv[g];
          ni = cc * oiv[g] - ss * prv[g];
        } else {                         // [[c,-s],[s,c]]
          float sg = bit ? ss : -ss;
          nr = cc * orv[g] + sg * prv[g];
          ni = cc * oiv[g] + sg * piv[g];
        }
        MRe[g * 256 + mr * 16 + mc] = nr;
        MIm[g * 256 + mr * 16 + mc] = ni;
      }
      __syncthreads();
    };
    auto xx_step = [&](int b) {          // new = c*own + (-i s)*partner(rows ^ mask)
      int mask = 3 << (2 - b);
      int rp = mr ^ mask;
      float orv[3], oiv[3], prv[3], piv[3];
#pragma unroll
      for (int g = 0; g < 3; ++g) {
        orv[g] = MRe[g * 256 + mr * 16 + mc];  oiv[g] = MIm[g * 256 + mr * 16 + mc];
        prv[g] = MRe[g * 256 + rp * 16 + mc];  piv[g] = MIm[g * 256 + rp * 16 + mc];
      }
      __syncthreads();
#pragma unroll
      for (int g = 0; g < 3; ++g) {
        float cc = entC[4 * g + b], ss = entS[4 * g + b];
        MRe[g * 256 + mr * 16 + mc] = cc * orv[g] + ss * piv[g];
        MIm[g * 256 + mr * 16 + mc] = cc * oiv[g] - ss * prv[g];
      }
      __syncthreads();
    };
    auto zz_step = [&](int b) {          // diagonal row phase e^{-+ i t/2}
      int b1 = (mr >> (3 - b)) & 1, b2 = (mr >> (2 - b)) & 1;
#pragma unroll
      for (int g = 0; g < 3; ++g) {
        float cc = entC[4 * g + b], ss = entS[4 * g + b];
        float sg = (b1 == b2) ? ss : -ss;
        float orv = MRe[g * 256 + mr * 16 + mc], oiv = MIm[g * 256 + mr * 16 + mc];
        MRe[g * 256 + mr * 16 + mc] = cc * orv + sg * oiv;
        MIm[g * 256 + mr * 16 + mc] = cc * oiv - sg * orv;
      }
      __syncthreads();
    };
    rot_step(0, true);
    xx_step(0);
    rot_step(1, false);
    zz_step(1);
    rot_step(2, true);
    xx_step(2);
    rot_step(3, false);

    // Per-layer pass matrices via single-wave WMMA matmuls (wave-uniform branches):
    //   P1_par = G1c x T_par ; P2 = K2 x G2c (pre-applies next layer's U2; last layer: G2c).
    if (wave == 0) {
      mm16_wave(P1Re,       P1Im,       MRe + 256, MIm + 256, TRe,       TIm,       lane);
    } else if (wave == 1) {
      mm16_wave(P1Re + 256, P1Im + 256, MRe + 256, MIm + 256, TRe + 256, TIm + 256, lane);
    } else if (wave == 2) {
      if (l < LAYERS - 1) {
        mm16_wave(P2Re, P2Im, KRe + 512, KIm + 512, MRe + 512, MIm + 512, lane);
      } else {
#pragma unroll
        for (int q = 0; q < 8; ++q) {
          P2Re[lane + 32 * q] = MRe[512 + lane + 32 * q];
          P2Im[lane + 32 * q] = MIm[512 + lane + 32 * q];
        }
      }
    }
    __syncthreads();

    // Three full-state passes: (G0c*U0), (G1c*ZZ34*U1 via per-parity A), (U2'*G2c with ZZ78 fold)
    apply_group(sAmp, MRe, MIm, 0, tid, false, 0.0f, 0.0f);
    __syncthreads();
    {
      const float* ar = (wave & 1) ? (P1Re + 256) : P1Re;
      const float* ai = (wave & 1) ? (P1Im + 256) : P1Im;
      apply_group(sAmp, ar, ai, 1, tid, false, 0.0f, 0.0f);
    }
    __syncthreads();
    apply_group(sAmp, P2Re, P2Im, 2, tid, true, entC[7], entS[7]);
    __syncthreads();
  }

  // <Z_i> = sum_k |amp_k|^2 * (1 - 2*bit_i(k))
  float acc[NQ];
#pragma unroll
  for (int i = 0; i < NQ; ++i) acc[i] = 0.0f;
#pragma unroll
  for (int k = 0; k < DIM / TPB; ++k) {
    int idx = tid + TPB * k;
    v2f a = sAmp[sp(idx)];
    float p = a[0] * a[0] + a[1] * a[1];
#pragma unroll
    for (int i = 0; i < NQ; ++i)
      acc[i] += ((idx >> (11 - i)) & 1) ? -p : p;
  }
#pragma unroll
  for (int off = 16; off >= 1; off >>= 1)
#pragma unroll
    for (int i = 0; i < NQ; ++i)
      acc[i] += __shfl_xor(acc[i], off, 32);
  if (lane == 0)
#pragma unroll
    for (int i = 0; i < NQ; ++i) red[wave * NQ + i] = acc[i];
  __syncthreads();
  if (tid < NQ) {
    float s = 0.0f;
#pragma unroll
    for (int w = 0; w < 8; ++w) s += red[w * NQ + tid];
    out[blk * NQ + tid] = s;
  }
}

extern "C" void kernel_launch(void* const* d_in, const int* in_sizes, int n_in,
                              void* d_out, int out_size, void* d_ws, size_t ws_size,
                              hipStream_t stream) {
  (void)n_in; (void)out_size; (void)d_ws; (void)ws_size;
  const float* feats = (const float*)d_in[0];
  const float* wrot  = (const float*)d_in[1];
  const float* wenta = (const float*)d_in[2];
  float* out = (float*)d_out;
  const int batch = in_sizes[0] / FEATS_PER;  // 2048
  qsim_kernel<<<batch, TPB, 0, stream>>>(feats, wrot, wenta, out);
}